// DetectionLoss_12910671692545
// MI455X (gfx1250) — compile-verified
//
#include <hip/hip_runtime.h>
#include <hip/hip_bf16.h>

// ---------------- problem constants (match reference) ----------------
#define DD 48
#define HH 48
#define WW 48
#define AA 3
#define GG 8
#define BB 4
#define APS (DD*HH*WW*AA)          // anchors per sample = 331776
#define TOTALN (BB*APS)            // 1327104
#define BPS (APS/256)              // blocks per sample = 1296
#define BLOCKS_MAIN (TOTALN/256)   // 5184
#define NBINS 4096
#define KNEG 800u
#define TH_POS_C 0.5f
#define TH_NEG_C 0.02f
#define POS_W 5.0f
#define REG_W 0.5f

// ---------------- ws layout (dwords) ----------------
// hist : uint [BB*NBINS]            @ 0
// ust  : uint [BB*8]                @ BB*NBINS
//        [0]=npos [1]=nneg [2]=K_eff [3]=active [4]=selbits [5]=prefix [6]=T [7]=count_gt
// fst  : float[BB*4]                [0]=pos_sum [1]=reg_sum [2]=sum_gt
// ppos/preg/p4 : float[BLOCKS_MAIN] per-block partials (deterministic reduce)
// keys : uint[TOTALN]               OPTIONAL (only if ws_size permits): key or 0
#define INIT_DWORDS (BB*NBINS + BB*8)
#define BASE_DWORDS (BB*NBINS + BB*8 + BB*4 + 3*BLOCKS_MAIN)

__device__ __forceinline__ float softplusf(float x) {
    // bce(x,0) = max(x,0) + log1p(exp(-|x|))
    return fmaxf(x, 0.f) + log1pf(expf(-fabsf(x)));
}
__device__ __forceinline__ float smoothl1(float x) {
    float ax = fabsf(x);
    return ax < 1.f ? 0.5f * x * x : ax - 0.5f;
}
__device__ __forceinline__ unsigned key_of(float conf) {
    unsigned u = __float_as_uint(conf);
    return (u & 0x80000000u) ? ~u : (u | 0x80000000u);  // order-preserving bijection
}
__device__ __forceinline__ float unkey(unsigned k) {
    unsigned u = (k & 0x80000000u) ? (k ^ 0x80000000u) : ~k;
    return __uint_as_float(u);
}
__device__ __forceinline__ void geom_of(int r, int& a, float& cx, float& cy, float& cz) {
    a = r % AA; r /= AA;
    int w = r % WW; r /= WW;
    int h = r % HH; int d = r / HH;
    cx = (float)w * 4.f + 2.f;
    cy = (float)h * 4.f + 2.f;
    cz = (float)d * 4.f + 2.f;
}
__device__ __forceinline__ float max_iou8(const float* sg, float cx, float cy, float cz,
                                          float anchor, int* argmax) {
    float r1 = 0.5f * anchor;
    float av = anchor * anchor * anchor;
    float best = -1.f; int bi = 0;
#pragma unroll
    for (int g = 0; g < GG; ++g) {
        float gx = sg[g*4+0], gy = sg[g*4+1], gz = sg[g*4+2], gd = sg[g*4+3];
        float r2 = 0.5f * gd;
        float ix = fmaxf(fminf(cx + r1, gx + r2) - fmaxf(cx - r1, gx - r2), 0.f);
        float iy = fmaxf(fminf(cy + r1, gy + r2) - fmaxf(cy - r1, gy - r2), 0.f);
        float iz = fmaxf(fminf(cz + r1, gz + r2) - fmaxf(cz - r1, gz - r2), 0.f);
        float inter = ix * iy * iz;
        float uni = av + gd * gd * gd - inter;
        float iou = inter / (uni + 1e-6f);
        if (iou > best) { best = iou; bi = g; }  // first-max == jnp.argmax
    }
    if (argmax) *argmax = bi;
    return best;
}

// ---------------- kernels ----------------
__global__ __launch_bounds__(256) void k_init(unsigned* ws) {
    int i = blockIdx.x * 256 + threadIdx.x;
    if (i < INIT_DWORDS) ws[i] = 0u;
}

// Pass 1: pos/neg classification, pos & reg partial sums, level-0 histogram,
// and (optionally) the per-anchor key array for the fast select passes.
__global__ __launch_bounds__(256) void k_main(const float* __restrict__ pred,
                                              const float* __restrict__ tgt,
                                              unsigned* __restrict__ hist,
                                              unsigned* __restrict__ ust,
                                              float* __restrict__ ppos,
                                              float* __restrict__ preg,
                                              unsigned* __restrict__ keys) {
    __shared__ float sg[GG * 4];
    __shared__ float sp[256], sr[256];
    __shared__ unsigned sn[256];
    int tid = threadIdx.x;
    int n   = blockIdx.x * 256 + tid;
    int b   = n / APS;                       // uniform per block (APS % 256 == 0)
    if (tid < GG * 4) sg[tid] = tgt[b * GG * 4 + tid];
    __syncthreads();

    int r = n % APS;
    int a; float cx, cy, cz;
    geom_of(r, a, cx, cy, cz);
    float anchor = (a == 0) ? 5.f : ((a == 1) ? 10.f : 20.f);

    int bi;
    float best = max_iou8(sg, cx, cy, cz, anchor, &bi);
    bool pos = best > TH_POS_C;
    bool neg = best < TH_NEG_C;

    const float* p = pred + (size_t)n * 5;
    float conf = p[0];

    float lp = 0.f, lr = 0.f; unsigned np = 0u;
    if (pos) {
        np = 1u;
        lp = softplusf(-conf);               // bce(conf, 1)
        float gx = sg[bi*4+0], gy = sg[bi*4+1], gz = sg[bi*4+2], gd = sg[bi*4+3];
        float tx = (gx - cx) / anchor;
        float ty = (gy - cy) / anchor;
        float tz = (gz - cz) / anchor;
        float td = logf(gd / anchor);
        lr = smoothl1(p[1] - tx) + smoothl1(p[2] - ty)
           + smoothl1(p[3] - tz) + smoothl1(p[4] - td);
    }
    unsigned key = neg ? key_of(conf) : 0u;
    if (keys) keys[n] = key;                 // 0 == "not a negative anchor"
    if (neg) atomicAdd(&hist[(unsigned)b * NBINS + (key >> 20)], 1u);

    // deterministic block reduction -> per-block partials (no float atomics)
    sp[tid] = lp; sr[tid] = lr; sn[tid] = np;
    __syncthreads();
    for (int s = 128; s > 0; s >>= 1) {
        if (tid < s) { sp[tid] += sp[tid+s]; sr[tid] += sr[tid+s]; sn[tid] += sn[tid+s]; }
        __syncthreads();
    }
    if (tid == 0) {
        ppos[blockIdx.x] = sp[0];
        preg[blockIdx.x] = sr[0];
        if (sn[0]) atomicAdd(&ust[b * 8 + 0], sn[0]);   // integer atomic: order-independent
    }
}

// ---- parallel radix-select scans: one block per sample, two-level chunk scan ----
__global__ __launch_bounds__(256) void k_scan1(unsigned* __restrict__ hist,
                                               unsigned* __restrict__ ust) {
    __shared__ unsigned cs[256];
    int b = blockIdx.x, t = threadIdx.x;
    unsigned* h = hist + b * NBINS;
    unsigned sum = 0;
#pragma unroll
    for (int i = 0; i < 16; ++i) sum += h[t * 16 + i];
    cs[t] = sum;
    __syncthreads();
    if (t == 0) {
        unsigned nneg = 0;
        for (int i = 0; i < 256; ++i) nneg += cs[i];
        unsigned K = nneg < KNEG ? nneg : KNEG;
        ust[b*8+1] = nneg; ust[b*8+2] = K;
        if (K == 0) {
            ust[b*8+3] = 0u; ust[b*8+6] = 0xFFFFFFFFu; ust[b*8+7] = 0u;
        } else {
            unsigned cum = 0; int chunk = 255;
            for (; chunk > 0; --chunk) {
                unsigned c = cs[chunk];
                if (cum + c >= K) break;
                cum += c;
            }
            unsigned sel = 0, pre = 0;
            for (int bin = chunk * 16 + 15; bin >= chunk * 16; --bin) {
                unsigned c = h[bin];
                if (cum + c >= K) { sel = (unsigned)bin; pre = cum; break; }
                cum += c;
            }
            ust[b*8+3] = 1u; ust[b*8+4] = sel << 20; ust[b*8+5] = pre;
        }
    }
    __syncthreads();                       // all reads done before re-zero
#pragma unroll
    for (int i = 0; i < 16; ++i) h[t * 16 + i] = 0u;
}

__global__ __launch_bounds__(256) void k_scan2(unsigned* __restrict__ hist,
                                               unsigned* __restrict__ ust) {
    __shared__ unsigned cs[256];
    int b = blockIdx.x, t = threadIdx.x;
    unsigned* h = hist + b * NBINS;
    unsigned sum = 0;
#pragma unroll
    for (int i = 0; i < 16; ++i) sum += h[t * 16 + i];
    cs[t] = sum;
    __syncthreads();
    if (t == 0 && ust[b*8+3]) {
        unsigned K = ust[b*8+2], cum = ust[b*8+5], sel = ust[b*8+4];
        int chunk = 255;
        for (; chunk > 0; --chunk) {
            unsigned c = cs[chunk];
            if (cum + c >= K) break;
            cum += c;
        }
        for (int bin = chunk * 16 + 15; bin >= chunk * 16; --bin) {
            unsigned c = h[bin];
            if (cum + c >= K) { ust[b*8+4] = sel | ((unsigned)bin << 8); ust[b*8+5] = cum; break; }
            cum += c;
        }
    }
    __syncthreads();
#pragma unroll
    for (int i = 0; i < 16; ++i) h[t * 16 + i] = 0u;
}

__global__ __launch_bounds__(256) void k_scan3(unsigned* __restrict__ hist,
                                               unsigned* __restrict__ ust) {
    __shared__ unsigned cs[256];
    int b = blockIdx.x, t = threadIdx.x;
    cs[t] = hist[b * NBINS + t];           // only 256 bins used at this level
    __syncthreads();
    if (t == 0 && ust[b*8+3]) {
        unsigned K = ust[b*8+2], cum = ust[b*8+5], sel = ust[b*8+4];
        for (int bin = 255; bin >= 0; --bin) {
            unsigned c = cs[bin];
            if (cum + c >= K) { ust[b*8+6] = sel | (unsigned)bin; ust[b*8+7] = cum; break; }
            cum += c;
        }
    }
}

// phase 2: mid 12 bits; phase 3: low 8 bits.  Fast path streams keys[] only;
// fallback recomputes IoU + reloads conf when ws has no room for keys[].
__global__ __launch_bounds__(256) void k_select_hist(const float* __restrict__ pred,
                                                     const float* __restrict__ tgt,
                                                     const unsigned* __restrict__ keys,
                                                     unsigned* __restrict__ hist,
                                                     const unsigned* __restrict__ ust,
                                                     int phase) {
    __shared__ float sg[GG * 4];
    int tid = threadIdx.x;
    int n   = blockIdx.x * 256 + tid;
    int b   = n / APS;
    if (!keys && tid < GG * 4) sg[tid] = tgt[b * GG * 4 + tid];
    __syncthreads();
    if (!ust[b*8+3]) return;                      // uniform per block

    unsigned key;
    if (keys) {
        key = keys[n];
        if (!key) return;
    } else {
        __builtin_prefetch(pred + (size_t)n * 5, 0, 0);  // hoist conf over IoU recompute
        int r = n % APS;
        int a; float cx, cy, cz;
        geom_of(r, a, cx, cy, cz);
        float anchor = (a == 0) ? 5.f : ((a == 1) ? 10.f : 20.f);
        float best = max_iou8(sg, cx, cy, cz, anchor, nullptr);
        if (!(best < TH_NEG_C)) return;
        key = key_of(pred[(size_t)n * 5]);
    }
    unsigned sel = ust[b*8+4];
    if (phase == 2) {
        if ((key & 0xFFF00000u) == sel)
            atomicAdd(&hist[(unsigned)b * NBINS + ((key >> 8) & 0xFFFu)], 1u);
    } else {
        if ((key & 0xFFFFFF00u) == sel)
            atomicAdd(&hist[(unsigned)b * NBINS + (key & 0xFFu)], 1u);
    }
}

// Final pass: sum softplus(conf) over keys strictly above threshold.
__global__ __launch_bounds__(256) void k_pass4(const float* __restrict__ pred,
                                               const float* __restrict__ tgt,
                                               const unsigned* __restrict__ keys,
                                               const unsigned* __restrict__ ust,
                                               float* __restrict__ p4) {
    __shared__ float sg[GG * 4];
    __shared__ float ss[256];
    int tid = threadIdx.x;
    int n   = blockIdx.x * 256 + tid;
    int b   = n / APS;
    if (!keys && tid < GG * 4) sg[tid] = tgt[b * GG * 4 + tid];
    __syncthreads();

    unsigned T = ust[b*8+6];
    float v = 0.f;
    if (keys) {
        unsigned key = keys[n];
        if (key && key > T) v = softplusf(unkey(key));  // conf recovered exactly
    } else {
        __builtin_prefetch(pred + (size_t)n * 5, 0, 0);
        int r = n % APS;
        int a; float cx, cy, cz;
        geom_of(r, a, cx, cy, cz);
        float anchor = (a == 0) ? 5.f : ((a == 1) ? 10.f : 20.f);
        float best = max_iou8(sg, cx, cy, cz, anchor, nullptr);
        if (best < TH_NEG_C) {
            float conf = pred[(size_t)n * 5];
            if (key_of(conf) > T) v = softplusf(conf);
        }
    }
    ss[tid] = v;
    __syncthreads();
    for (int s = 128; s > 0; s >>= 1) {
        if (tid < s) ss[tid] += ss[tid + s];
        __syncthreads();
    }
    if (tid == 0) p4[blockIdx.x] = ss[0];
}

// Deterministic fixed-order reduce of per-block partials (one block per sample)
__global__ __launch_bounds__(256) void k_reduce(const float* __restrict__ src,
                                                float* __restrict__ fst, int slot) {
    __shared__ float ss[256];
    int b = blockIdx.x, tid = threadIdx.x;
    float acc = 0.f;
    for (int i = tid; i < BPS; i += 256) acc += src[b * BPS + i];
    ss[tid] = acc;
    __syncthreads();
    for (int s = 128; s > 0; s >>= 1) {
        if (tid < s) ss[tid] += ss[tid + s];
        __syncthreads();
    }
    if (tid == 0) fst[b * 4 + slot] = ss[0];
}

__global__ void k_final(const unsigned* __restrict__ ust,
                        const float* __restrict__ fst,
                        float* __restrict__ out) {
    if (threadIdx.x != 0 || blockIdx.x != 0) return;
    float cls_sum = 0.f, reg_sum = 0.f, npos_sum = 0.f;
    for (int b = 0; b < BB; ++b) {
        float npos    = (float)ust[b*8+0];
        unsigned K    = ust[b*8+2];
        unsigned T    = ust[b*8+6];
        unsigned cgt  = ust[b*8+7];
        float pos_s   = fst[b*4+0];
        float reg_s   = fst[b*4+1];
        float sgt     = fst[b*4+2];

        float pos_loss = (npos > 0.f) ? POS_W * pos_s / fmaxf(npos, 1.f) : 0.f;
        float neg_loss = 0.f;
        if (K > 0u) {
            float tv = softplusf(unkey(T));                 // threshold value (ties)
            float s  = sgt + (float)(K - cgt) * tv;         // exact top-K sum
            neg_loss = s / fmaxf((float)K, 1.f);
        }
        float reg_loss = (npos > 0.f) ? reg_s / fmaxf(npos * 4.f, 1.f) : 0.f;

        cls_sum  += pos_loss + neg_loss;
        reg_sum  += reg_loss;
        npos_sum += npos;
    }
    float cls = cls_sum / (float)BB;
    float reg = reg_sum / (float)BB;
    out[0] = cls + REG_W * reg;
    out[1] = cls;
    out[2] = reg;
    out[3] = npos_sum;
}

// ---------------- host ----------------
extern "C" void kernel_launch(void* const* d_in, const int* in_sizes, int n_in,
                              void* d_out, int out_size, void* d_ws, size_t ws_size,
                              hipStream_t stream) {
    const float* pred = (const float*)d_in[0];   // (4,48,48,48,3,5) f32
    const float* tgt  = (const float*)d_in[1];   // (4,8,4) f32
    float* out = (float*)d_out;                  // [total, cls, reg, npos]

    unsigned* hist = (unsigned*)d_ws;                    // [BB*NBINS]
    unsigned* ust  = hist + BB * NBINS;                  // [BB*8]
    float*    fst  = (float*)(ust + BB * 8);             // [BB*4]
    float*    ppos = fst + BB * 4;                       // [BLOCKS_MAIN]
    float*    preg = ppos + BLOCKS_MAIN;                 // [BLOCKS_MAIN]
    float*    p4   = preg + BLOCKS_MAIN;                 // [BLOCKS_MAIN]
    // optional key cache: select passes stream 5.3MB of keys instead of
    // re-reading 26.5MB of pred + recomputing IoU (ws_size is call-invariant,
    // so the chosen path — and the output — is deterministic)
    unsigned* keys = (ws_size >= ((size_t)BASE_DWORDS + (size_t)TOTALN) * 4u)
                   ? ((unsigned*)d_ws + BASE_DWORDS) : nullptr;

    k_init<<<(INIT_DWORDS + 255) / 256, 256, 0, stream>>>((unsigned*)d_ws);
    k_main<<<BLOCKS_MAIN, 256, 0, stream>>>(pred, tgt, hist, ust, ppos, preg, keys);
    k_reduce<<<BB, 256, 0, stream>>>(ppos, fst, 0);
    k_reduce<<<BB, 256, 0, stream>>>(preg, fst, 1);
    k_scan1<<<BB, 256, 0, stream>>>(hist, ust);
    k_select_hist<<<BLOCKS_MAIN, 256, 0, stream>>>(pred, tgt, keys, hist, ust, 2);
    k_scan2<<<BB, 256, 0, stream>>>(hist, ust);
    k_select_hist<<<BLOCKS_MAIN, 256, 0, stream>>>(pred, tgt, keys, hist, ust, 3);
    k_scan3<<<BB, 256, 0, stream>>>(hist, ust);
    k_pass4<<<BLOCKS_MAIN, 256, 0, stream>>>(pred, tgt, keys, ust, p4);
    k_reduce<<<BB, 256, 0, stream>>>(p4, fst, 2);
    k_final<<<1, 1, 0, stream>>>(ust, fst, out);
}